// ROIAlign_65584150610282
// MI455X (gfx1250) — compile-verified
//
#include <hip/hip_runtime.h>
#include <stdint.h>

#define NCH 256

// ---------------------------------------------------------------------------
// Shared geometry helpers (exactly mirror the reference math, fp32)
// ---------------------------------------------------------------------------
__device__ __forceinline__ int iclamp(int v, int lo, int hi) {
  return v < lo ? lo : (v > hi ? hi : v);
}

__device__ __forceinline__ void roi_geom(float px1, float py1, float px2, float py2,
                                         int& lvl, int& W, float& bx1, float& by1,
                                         float& wu, float& hu) {
  float pw = px2 - px1, ph = py2 - py1;
  float lf = floorf(2.0f + log2f(sqrtf(pw * ph) / 224.0f));
  lf = fminf(fmaxf(lf, 0.0f), 3.0f);
  lvl = (int)lf;
  const float SC[4] = {0.25f, 0.125f, 0.0625f, 0.03125f};
  const int   SZ[4] = {256, 128, 64, 32};
  float sc = SC[lvl];
  W = SZ[lvl];
  bx1 = px1 * sc;  by1 = py1 * sc;
  float bx2 = px2 * sc, by2 = py2 * sc;
  wu = ((bx2 - bx1) / 7.0f) * 0.5f;   // w_unit = w_stride / S
  hu = ((by2 - by1) / 7.0f) * 0.5f;
}

// Tap indices + weights for one axis. When clipping collapses the index pair,
// the reference's weighted sum cancels to (numerically negligible) zero, so we
// flag invalid and emit exactly 0 for that sample.
__device__ __forceinline__ void axis_tap(float x, int W, int& x0, int& ok,
                                         float& lx, float& hx) {
  float fx = floorf(x);
  int ix = (int)fx;
  x0 = iclamp(ix, 0, W - 1);
  int x1 = iclamp(ix + 1, 0, W - 1);
  ok = (x1 == x0 + 1) ? 1 : 0;
  lx = x - (float)x0;
  hx = (float)x1 - x;
}

// ---------------------------------------------------------------------------
// TDM: load a 2-row x (2*NCH)-element f32 tile (2 adjacent pixels, all
// channels, 2 adjacent rows) from the channels-last feature into LDS.
// D# packing per CDNA5 ISA ch.8 (group0: count/lds_addr/global_addr/type=2,
// group1: data_size=4B, tile_dim0=512, tile_dim1=2, dim0_stride=W*NCH).
// Tensor dims set huge (2^30) so OOB-zeroing never triggers; out-of-bounds
// tiles carry zero bilinear weight and only touch padded workspace.
// ---------------------------------------------------------------------------
typedef unsigned int u32x4 __attribute__((ext_vector_type(4)));
typedef int          i32x4 __attribute__((ext_vector_type(4)));
typedef int          i32x8 __attribute__((ext_vector_type(8)));

__device__ __forceinline__ void tdm_load_tile(uint64_t gaddr, uint32_t lds_addr,
                                              int stride_elems) {
#if defined(__AMDGCN__)
  u32x4 g0;
  g0.x = (unsigned)__builtin_amdgcn_readfirstlane(1);                 // count=1, user mode
  g0.y = (unsigned)__builtin_amdgcn_readfirstlane((int)lds_addr);     // lds_addr (bytes)
  g0.z = (unsigned)__builtin_amdgcn_readfirstlane((int)(uint32_t)gaddr);
  g0.w = (unsigned)__builtin_amdgcn_readfirstlane(
      (int)(((uint32_t)(gaddr >> 32) & 0x01FFFFFFu) | (2u << 30)));   // addr[56:32] | type=2
  i32x8 g1;
  g1[0] = 0x00020000;              // workgroup_mask=0, data_size=2 (4 bytes)
  g1[1] = 0;                       // tensor_dim0 lo16 (dim0 = 1<<30)
  g1[2] = 0x00004000;              // tensor_dim0 hi16 ; tensor_dim1 lo16 (dim1 = 1<<30)
  g1[3] = (512 << 16) | 0x4000;    // tensor_dim1 hi16 ; tile_dim0 = 512 elems
  g1[4] = 2;                       // tile_dim1 = 2, tile_dim2 = 0
  g1[5] = __builtin_amdgcn_readfirstlane(stride_elems); // tensor_dim0_stride lo32
  g1[6] = 0;
  g1[7] = 0;
  i32x4 gz = {0, 0, 0, 0};
#if __clang_major__ >= 23
  i32x8 gz8 = {0, 0, 0, 0, 0, 0, 0, 0};
  __builtin_amdgcn_tensor_load_to_lds(g0, g1, gz, gz, gz8, 0);
#else
  __builtin_amdgcn_tensor_load_to_lds(g0, g1, gz, gz, 0);
#endif
#endif
}

// ---------------------------------------------------------------------------
// Kernel 1: per-level transpose (C, P) -> (P, C), P = H*W (always mult of 32)
// ---------------------------------------------------------------------------
__global__ __launch_bounds__(256) void transpose_cp(const float* __restrict__ in,
                                                    float* __restrict__ out, int P) {
  __shared__ float tile[32][33];
  int p0 = blockIdx.x * 32;
  int c0 = blockIdx.y * 32;
  int tx = threadIdx.x, ty = threadIdx.y;
#pragma unroll
  for (int j = 0; j < 32; j += 8)
    tile[ty + j][tx] = in[(size_t)(c0 + ty + j) * P + p0 + tx];
  __syncthreads();
#pragma unroll
  for (int j = 0; j < 32; j += 8)
    out[(size_t)(p0 + ty + j) * NCH + c0 + tx] = tile[tx][ty + j];
}

// ---------------------------------------------------------------------------
// Kernel 2: one block per ROI. Wave 0 drives a 4-deep TDM pipeline staging
// 4KB tap-tiles into LDS; 2 samples are retired per barrier pair. All 8 waves
// (thread = channel) do the bilinear blend + 2x2 max.
// ---------------------------------------------------------------------------
__global__ __launch_bounds__(256) void roi_align_tdm(const float* __restrict__ prop,
                                                     const float* __restrict__ ft,
                                                     float* __restrict__ out) {
  __shared__ __align__(16) float tbuf[4][4 * NCH];
  // Precomputed per-axis tap tables (14 grid positions each)
  __shared__ float sxl[14], sxh[14], syl[14], syh[14];
  __shared__ int   sx0[14], sy0[14], sxok[14], syok[14];

  int n = blockIdx.x;
  float4 p = reinterpret_cast<const float4*>(prop)[n];
  int lvl, W; float bx1, by1, wu, hu;
  roi_geom(p.x, p.y, p.z, p.w, lvl, W, bx1, by1, wu, hu);

  const long long OFFS[4] = {0LL, 16777216LL, 20971520LL, 22020096LL};  // floats
  uint64_t base = (uint64_t)(uintptr_t)ft + (uint64_t)(OFFS[lvl] * 4ll);
  uint32_t lds0 = (uint32_t)(uintptr_t)(&tbuf[0][0]);   // flat LDS addr[31:0] = offset
  int strideE = W * NCH;

  int c = threadIdx.x;
  bool issuer = (threadIdx.x < 32);                     // wave 0 only (uniform per wave)

  // ---- precompute tap tables --------------------------------------------
  if (threadIdx.x < 14) {
    int g = threadIdx.x;
    float x = bx1 + ((float)g + 0.5f) * wu;
    axis_tap(x, W, sx0[g], sxok[g], sxl[g], sxh[g]);
  } else if (threadIdx.x >= 32 && threadIdx.x < 46) {
    int g = threadIdx.x - 32;
    float y = by1 + ((float)g + 0.5f) * hu;
    axis_tap(y, W, sy0[g], syok[g], syl[g], syh[g]);
  }
  __syncthreads();

  // ---- prologue: fill the 4-deep pipeline (samples 0..3) ----------------
  if (issuer) {
#pragma unroll
    for (int s = 0; s < 4; ++s) {
      int cell = s >> 2, sub = s & 3;
      int gy = (cell / 7) * 2 + (sub >> 1);
      int gx = (cell % 7) * 2 + (sub & 1);
      uint64_t ga = base +
          ((uint64_t)((int64_t)sy0[gy] * W + sx0[gx]) * NCH) * 4ull;
      tdm_load_tile(ga, lds0 + (uint32_t)(s & 3) * (4u * NCH * 4u), strideE);
    }
  }

  float* op = out + ((size_t)n * NCH + c) * 49;
  float m = 0.0f;

  // 98 pipeline steps, 2 samples each
  for (int t = 0; t < 98; ++t) {
    if (issuer) {
      if (t <= 96) __builtin_amdgcn_s_wait_tensorcnt(2);   // samples 2t,2t+1 done
      else         __builtin_amdgcn_s_wait_tensorcnt(0);   // drain
    }
    __syncthreads();   // publish LDS tiles to all waves

#pragma unroll
    for (int k = 0; k < 2; ++k) {
      int s = 2 * t + k;
      int cell = s >> 2, sub = s & 3;
      int gy = (cell / 7) * 2 + (sub >> 1);
      int gx = (cell % 7) * 2 + (sub & 1);
      float val = 0.0f;
      if (sxok[gx] & syok[gy]) {
        float lx = sxl[gx], hx = sxh[gx], ly = syl[gy], hy = syh[gy];
        const float* T = tbuf[s & 3];
        float v00 = T[c];             // (y0, x0)
        float v01 = T[NCH + c];       // (y0, x1)
        float v10 = T[2 * NCH + c];   // (y1, x0)
        float v11 = T[3 * NCH + c];   // (y1, x1)
        val = (((hx * hy) * v00 + (hx * ly) * v10) + (lx * hy) * v01) + (lx * ly) * v11;
      }
      m = (sub == 0) ? val : fmaxf(m, val);
      if (sub == 3) op[cell] = m;
    }

    __syncthreads();   // all readers done with both tiles before refill

    if (issuer) {
#pragma unroll
      for (int k = 0; k < 2; ++k) {
        int s2 = 2 * t + 4 + k;
        if (s2 < 196) {
          int cell2 = s2 >> 2, sub2 = s2 & 3;
          int gy2 = (cell2 / 7) * 2 + (sub2 >> 1);
          int gx2 = (cell2 % 7) * 2 + (sub2 & 1);
          uint64_t ga = base +
              ((uint64_t)((int64_t)sy0[gy2] * W + sx0[gx2]) * NCH) * 4ull;
          tdm_load_tile(ga, lds0 + (uint32_t)(s2 & 3) * (4u * NCH * 4u), strideE);
        }
      }
    }
  }
}

// ---------------------------------------------------------------------------
// Fallback: direct gather from the original (C, H, W) layout (no workspace).
// ---------------------------------------------------------------------------
__global__ __launch_bounds__(256) void roi_align_direct(
    const float* __restrict__ prop,
    const float* __restrict__ f2, const float* __restrict__ f3,
    const float* __restrict__ f4, const float* __restrict__ f5,
    float* __restrict__ out) {
  int n = blockIdx.x;
  float4 p = reinterpret_cast<const float4*>(prop)[n];
  int lvl, W; float bx1, by1, wu, hu;
  roi_geom(p.x, p.y, p.z, p.w, lvl, W, bx1, by1, wu, hu);
  const float* F = (lvl == 0) ? f2 : (lvl == 1) ? f3 : (lvl == 2) ? f4 : f5;

  int c = threadIdx.x;
  const float* Fc = F + (size_t)c * (size_t)W * (size_t)W;
  float* op = out + ((size_t)n * NCH + c) * 49;

  for (int cell = 0; cell < 49; ++cell) {
    int oy = cell / 7, ox = cell % 7;
    float m = 0.0f;
#pragma unroll
    for (int sub = 0; sub < 4; ++sub) {
      int gy = oy * 2 + (sub >> 1);
      int gx = ox * 2 + (sub & 1);
      float x = bx1 + ((float)gx + 0.5f) * wu;
      float y = by1 + ((float)gy + 0.5f) * hu;
      int x0, y0, okx, oky; float lx, hx, ly, hy;
      axis_tap(x, W, x0, okx, lx, hx);
      axis_tap(y, W, y0, oky, ly, hy);
      float val = 0.0f;
      if (okx & oky) {
        const float* r0 = Fc + (size_t)y0 * W + x0;
        float v00 = r0[0], v01 = r0[1];
        float v10 = r0[W], v11 = r0[W + 1];
        val = (((hx * hy) * v00 + (hx * ly) * v10) + (lx * hy) * v01) + (lx * ly) * v11;
      }
      m = (sub == 0) ? val : fmaxf(m, val);
    }
    op[cell] = m;
  }
}

// ---------------------------------------------------------------------------
extern "C" void kernel_launch(void* const* d_in, const int* in_sizes, int n_in,
                              void* d_out, int out_size, void* d_ws, size_t ws_size,
                              hipStream_t stream) {
  const float* f[4] = {(const float*)d_in[0], (const float*)d_in[1],
                       (const float*)d_in[2], (const float*)d_in[3]};
  const float* prop = (const float*)d_in[4];
  float* out = (float*)d_out;
  int N = in_sizes[4] / 4;

  const int SZ[4] = {256, 128, 64, 32};
  const size_t OFF[4] = {0, 16777216, 20971520, 22020096};     // floats
  const size_t TOT = 22282240;                                  // floats
  const size_t need = TOT * sizeof(float) + 65536;              // +slack for tile overrun

  if (d_ws != nullptr && ws_size >= need) {
    float* ft = (float*)d_ws;
    for (int l = 0; l < 4; ++l) {
      int P = SZ[l] * SZ[l];
      dim3 b(32, 8), g(P / 32, NCH / 32);
      transpose_cp<<<g, b, 0, stream>>>(f[l], ft + OFF[l], P);
    }
    roi_align_tdm<<<dim3(N), dim3(256), 0, stream>>>(prop, ft, out);
  } else {
    roi_align_direct<<<dim3(N), dim3(256), 0, stream>>>(prop, f[0], f[1], f[2], f[3], out);
  }
}